// MPSRegressor_77335181132055
// MI455X (gfx1250) — compile-verified
//
#include <hip/hip_runtime.h>
#include <hip/hip_bf16.h>
#include <stdint.h>

// ---------------------------------------------------------------------------
// MPS chain regression on MI455X (gfx1250), wave32 + V_WMMA_F32_16X16X4_F32.
// State kept transposed (W = v^T, bond x samples): lane == sample in both the
// WMMA B-operand and D-output, so per-sample phi scaling is lane-local.
// The C-layout -> B-layout bond permutation pi is baked into the repacked
// cores, so each step's D registers feed the next step's B operand directly
// (zero inter-step shuffles). Cores stream through LDS via the Tensor Data
// Mover, double buffered.
// ---------------------------------------------------------------------------

typedef float v2f __attribute__((ext_vector_type(2)));
typedef float v8f __attribute__((ext_vector_type(8)));
typedef unsigned int v4u __attribute__((ext_vector_type(4)));
typedef int v4i __attribute__((ext_vector_type(4)));
typedef int v8i __attribute__((ext_vector_type(8)));

#define B_TOTAL 32768
#define L_CHAIN 256
#define D_BOND 16
#define NSTEP 254                 // L-2 middle cores
#define TSTEP 8                   // steps per LDS chunk
#define NCHUNK ((NSTEP + TSTEP - 1) / TSTEP)   // 32 (last chunk = 6 steps)
#define STEP_FLOATS 512           // 2 (p) * 4 (K-chunks) * 32 lanes * 2 floats
#define BLOCK_THREADS 128
#define TILES_PER_WAVE 2
#define SAMPLES_PER_WAVE (TILES_PER_WAVE * 16)          // 32
#define SAMPLES_PER_BLOCK ((BLOCK_THREADS / 32) * SAMPLES_PER_WAVE)  // 128

// ---------------------------------------------------------------------------
// Prep: repack cores_mid[t][q][p][r] into the WMMA A-operand register image,
// with the K-columns pre-permuted by pi so that D registers can be reused
// directly as the next step's B operand.
//   B slot k_phys = 4c + i + 2h physically holds logical row
//   pi(k_phys) = 2c + i + 8h   (D register 2c+i, half h holds row (2c+i)+8h).
// A-layout position: chunk c, lane half hA, dword j -> k_phys = 4c + 2hA + j.
// ws flat: (((t*2 + p)*4 + c)*32 + L)*2 + j
// ---------------------------------------------------------------------------
__global__ void mps_prep_cores(const float* __restrict__ cores_mid,
                               float* __restrict__ ws) {
  int idx = blockIdx.x * 256 + threadIdx.x;
  if (idx >= NSTEP * STEP_FLOATS) return;
  int t  = idx >> 9;
  int r5 = idx & 511;
  int p  = r5 >> 8;
  int c  = (r5 >> 6) & 3;
  int ln = (r5 >> 1) & 31;
  int j  = r5 & 1;
  int k  = 4 * c + 2 * (ln >> 4) + j;                    // physical K slot
  int q  = 2 * (k >> 2) + (k & 1) + 8 * ((k >> 1) & 1);  // logical bond-in = pi(k)
  int m  = ln & 15;                                      // bond-out (M = r)
  ws[idx] = cores_mid[((t * D_BOND + q) * 2 + p) * D_BOND + m];
}

// Issue one TDM 1D copy: elems floats from gsrc -> LDS offset lds_off.
__device__ __forceinline__ void tdm_load_chunk(const float* gsrc,
                                               unsigned lds_off, int elems) {
  unsigned long long ga = (unsigned long long)(uintptr_t)gsrc;
  unsigned E = (unsigned)elems;               // <= 4096, fits tile_dim0
  v4u g0;
  g0.x = 1u;                                   // count = 1 valid descriptor
  g0.y = lds_off;                              // lds_addr (bytes)
  g0.z = (unsigned)(ga & 0xFFFFFFFFu);         // global_addr[31:0]
  g0.w = (unsigned)((ga >> 32) & 0x01FFFFFFu)  // global_addr[56:32]
         | 0x80000000u;                        // type = 2 ("image")
  v8i g1;
  g1[0] = (int)0x00020000u;                    // data_size = 2 (4 bytes)
  g1[1] = (int)((E & 0xFFFFu) << 16);          // tensor_dim0[15:0]
  g1[2] = (int)((E >> 16) | (1u << 16));       // tensor_dim0[31:16] | tensor_dim1=1
  g1[3] = (int)((E & 0xFFFFu) << 16);          // tile_dim0 = E
  g1[4] = 0;                                   // tile_dim1 = tile_dim2 = 0 (1D)
  g1[5] = (int)E;                              // tensor_dim0_stride
  g1[6] = 0;
  g1[7] = 0;
  v4i gz = {0, 0, 0, 0};
#if __has_include(<hip/amd_detail/amd_gfx1250_TDM.h>)
  v8i gz8 = {0, 0, 0, 0, 0, 0, 0, 0};
  __builtin_amdgcn_tensor_load_to_lds(g0, g1, gz, gz, gz8, 0);
#else
  __builtin_amdgcn_tensor_load_to_lds(g0, g1, gz, gz, 0);
#endif
}

__global__ __launch_bounds__(BLOCK_THREADS)
void mps_chain_kernel(const float* __restrict__ phi,
                      const float* __restrict__ core_first,
                      const float* __restrict__ ws_cores,
                      const float* __restrict__ core_last,
                      const float* __restrict__ out_bias,
                      float* __restrict__ out) {
  __shared__ float smem[2][TSTEP * STEP_FLOATS];   // 2 x 16 KB double buffer

  const int lane = threadIdx.x & 31;
  const int wave = threadIdx.x >> 5;
  const int col  = lane & 15;          // sample column within tile
  const bool lo  = lane < 16;
  const int sbase = blockIdx.x * SAMPLES_PER_BLOCK + wave * SAMPLES_PER_WAVE;

  const float2* phi2 = (const float2*)phi;   // phi2[b*L + l] = {phi0, phi1}

  // W is kept in C-layout register order: register s, half h <-> bond row s+8h.
  int qrow[8];
#pragma unroll
  for (int s = 0; s < 8; ++s)
    qrow[s] = s + (lo ? 0 : 8);

  // ---- W init: v0 = phi[:,0] @ core_first, transposed, C-layout order ----
  float f0[8], f1[8];
#pragma unroll
  for (int s = 0; s < 8; ++s) {
    f0[s] = core_first[qrow[s]];        // core_first[0][q]
    f1[s] = core_first[16 + qrow[s]];   // core_first[1][q]
  }
  float w[TILES_PER_WAVE][8];
  int samp[TILES_PER_WAVE];
#pragma unroll
  for (int tl = 0; tl < TILES_PER_WAVE; ++tl) {
    samp[tl] = sbase + tl * 16 + col;
    float2 ph = phi2[samp[tl] * L_CHAIN + 0];
#pragma unroll
    for (int s = 0; s < 8; ++s)
      w[tl][s] = ph.x * f0[s] + ph.y * f1[s];
  }

  // ---- prime TDM double buffer (wave 0 issues for the workgroup) ----
  if (wave == 0) {
    tdm_load_chunk(ws_cores, (unsigned)(uintptr_t)&smem[0][0],
                   TSTEP * STEP_FLOATS);
    tdm_load_chunk(ws_cores + TSTEP * STEP_FLOATS,
                   (unsigned)(uintptr_t)&smem[1][0], TSTEP * STEP_FLOATS);
  }

  int tglob = 0;
  for (int k = 0; k < NCHUNK; ++k) {
    if (wave == 0) {
      if (k == NCHUNK - 1) __builtin_amdgcn_s_wait_tensorcnt(0);
      else                 __builtin_amdgcn_s_wait_tensorcnt(1);
    }
    __syncthreads();

    const int steps = (k == NCHUNK - 1) ? (NSTEP - k * TSTEP) : TSTEP;
    const float* sbuf = &smem[k & 1][0];

    for (int sl = 0; sl < steps; ++sl, ++tglob) {
      const float* ab = sbuf + sl * STEP_FLOATS;
      v2f a0[4], a1[4];
#pragma unroll
      for (int c = 0; c < 4; ++c) {
        a0[c] = *(const v2f*)(ab + c * 64 + lane * 2);        // C0^T chunk (pi-permuted K)
        a1[c] = *(const v2f*)(ab + 256 + c * 64 + lane * 2);  // C1^T chunk (pi-permuted K)
      }
#pragma unroll
      for (int tl = 0; tl < TILES_PER_WAVE; ++tl) {
        float2 ph = phi2[samp[tl] * L_CHAIN + (1 + tglob)];
        v8f u0 = {0.f, 0.f, 0.f, 0.f, 0.f, 0.f, 0.f, 0.f};
        v8f u1 = {0.f, 0.f, 0.f, 0.f, 0.f, 0.f, 0.f, 0.f};
#pragma unroll
        for (int c = 0; c < 4; ++c) {
          v2f bop;
          bop.x = w[tl][2 * c];
          bop.y = w[tl][2 * c + 1];
          u0 = __builtin_amdgcn_wmma_f32_16x16x4_f32(
              false, a0[c], false, bop, (short)0, u0, false, false);
          u1 = __builtin_amdgcn_wmma_f32_16x16x4_f32(
              false, a1[c], false, bop, (short)0, u1, false, false);
        }
        // per-sample (per-lane) physical-index contraction; D registers become
        // next step's B operand directly (pi absorbed into the cores).
#pragma unroll
        for (int j = 0; j < 8; ++j)
          w[tl][j] = ph.x * u0[j] + ph.y * u1[j];
      }
    }

    __syncthreads();
    if ((wave == 0) && (k + 2 < NCHUNK)) {
      int tstart = (k + 2) * TSTEP;
      int e = ((tstart + TSTEP <= NSTEP) ? TSTEP : (NSTEP - tstart)) * STEP_FLOATS;
      tdm_load_chunk(ws_cores + tstart * STEP_FLOATS,
                     (unsigned)(uintptr_t)&smem[k & 1][0], e);
    }
  }

  // ---- close chain: out = sum_q v[q] * (cl[q][0]*phiL0 + cl[q][1]*phiL1) ----
  const float bias = out_bias[0];
#pragma unroll
  for (int tl = 0; tl < TILES_PER_WAVE; ++tl) {
    float2 ph = phi2[samp[tl] * L_CHAIN + (L_CHAIN - 1)];
    float t0 = 0.f, t1 = 0.f;
#pragma unroll
    for (int s = 0; s < 8; ++s) {
      t0 += w[tl][s] * core_last[2 * qrow[s]];
      t1 += w[tl][s] * core_last[2 * qrow[s] + 1];
    }
    t0 += __shfl_xor(t0, 16, 32);   // combine bond rows held by other half
    t1 += __shfl_xor(t1, 16, 32);
    float res = ph.x * t0 + ph.y * t1 + bias;
    if (lo) out[samp[tl]] = res;
  }
}

extern "C" void kernel_launch(void* const* d_in, const int* in_sizes, int n_in,
                              void* d_out, int out_size, void* d_ws, size_t ws_size,
                              hipStream_t stream) {
  const float* phi        = (const float*)d_in[0];
  const float* core_first = (const float*)d_in[1];
  const float* cores_mid  = (const float*)d_in[2];
  const float* core_last  = (const float*)d_in[3];
  const float* out_bias   = (const float*)d_in[4];
  float* out = (float*)d_out;
  float* ws  = (float*)d_ws;   // needs NSTEP*STEP_FLOATS*4 = 508 KB

  const int prep_elems = NSTEP * STEP_FLOATS;
  mps_prep_cores<<<(prep_elems + 255) / 256, 256, 0, stream>>>(cores_mid, ws);
  mps_chain_kernel<<<B_TOTAL / SAMPLES_PER_BLOCK, BLOCK_THREADS, 0, stream>>>(
      phi, core_first, ws, core_last, out_bias, out);
}